// RGATPolicy_18081812316681
// MI455X (gfx1250) — compile-verified
//
#include <hip/hip_runtime.h>
#include <cstddef>

#define HEADS 8
#define HC 256
#define NREL 3
#define NEG_SLOPE 0.2f

typedef float v2f __attribute__((ext_vector_type(2)));
typedef float v8f __attribute__((ext_vector_type(8)));

// ---- order-preserving float <-> uint for atomicMax-based segment max ----
__device__ __forceinline__ unsigned f2ord(float f) {
  unsigned u = __float_as_uint(f);
  return (u & 0x80000000u) ? ~u : (u | 0x80000000u);
}
__device__ __forceinline__ float ord2f(unsigned s) {
  unsigned u = (s & 0x80000000u) ? (s & 0x7FFFFFFFu) : ~s;
  return __uint_as_float(u);
}

__global__ void fill_u32_kernel(unsigned* __restrict__ p, unsigned v, long n) {
  long i = (long)blockIdx.x * blockDim.x + threadIdx.x;
  if (i < n) p[i] = v;
}

// QK[k][j] = q[k][j] (j<8) else k[k][j-8]; wee[h] = sum_c we[c]*e[c][h]
__global__ void build_qk_wee_kernel(const float* __restrict__ q, const float* __restrict__ kmat,
                                    const float* __restrict__ emat, const float* __restrict__ we,
                                    float* __restrict__ QK, float* __restrict__ wee) {
  int t = blockIdx.x * blockDim.x + threadIdx.x;
  if (t < HC * 16) {
    int row = t >> 4, col = t & 15;
    QK[t] = (col < HEADS) ? q[row * HEADS + col] : kmat[row * HEADS + (col - HEADS)];
  }
  if (t < HEADS) {
    float s = 0.f;
    for (int c = 0; c < HC; ++c) s += we[c] * emat[c * HEADS + t];
    wee[t] = s;
  }
}

// XW[n, r, :] = A[n, :] @ W[r]  -- fp32 WMMA, one wave computes 16 rows x 64 cols
// (4 N-tiles share one A fragment per K-step).  grid = (M/16, HC/64, NREL), block = 32
__global__ void gemm_xw_kernel(const float* __restrict__ A, const float* __restrict__ W,
                               float* __restrict__ XW, int K) {
  int lane = threadIdx.x;
  int half = lane >> 4, col = lane & 15;
  int mt = blockIdx.x, ng = blockIdx.y, r = blockIdx.z;
  const float* Arow  = A + (size_t)(mt * 16 + col) * K;               // row m = lane&15
  const float* Bbase = W + (size_t)r * K * HC + ng * 64 + col;        // col base
  v8f acc0 = {}, acc1 = {}, acc2 = {}, acc3 = {};
  for (int k0 = 0; k0 < K; k0 += 4) {
    int ka = k0 + 2 * half;                   // A frag: A[m][ka], A[m][ka+1]
    __builtin_prefetch(Arow + ka + 32, 0, 1);
    v2f a = { Arow[ka], Arow[ka + 1] };
    const float* brow0 = Bbase + (size_t)ka * HC;        // B row ka
    const float* brow1 = Bbase + (size_t)(ka + 1) * HC;  // B row ka+1
    v2f b0 = { brow0[0],  brow1[0]  };
    v2f b1 = { brow0[16], brow1[16] };
    v2f b2 = { brow0[32], brow1[32] };
    v2f b3 = { brow0[48], brow1[48] };
    acc0 = __builtin_amdgcn_wmma_f32_16x16x4_f32(false, a, false, b0, (short)0, acc0, false, false);
    acc1 = __builtin_amdgcn_wmma_f32_16x16x4_f32(false, a, false, b1, (short)0, acc1, false, false);
    acc2 = __builtin_amdgcn_wmma_f32_16x16x4_f32(false, a, false, b2, (short)0, acc2, false, false);
    acc3 = __builtin_amdgcn_wmma_f32_16x16x4_f32(false, a, false, b3, (short)0, acc3, false, false);
  }
#pragma unroll
  for (int v = 0; v < 8; ++v) {
    int m = v + 8 * half;                     // C layout: lane holds (M=v+8*half, N=lane&15)
    size_t o = ((size_t)(mt * 16 + m) * NREL + r) * HC + ng * 64 + col;
    XW[o]      = acc0[v];
    XW[o + 16] = acc1[v];
    XW[o + 32] = acc2[v];
    XW[o + 48] = acc3[v];
  }
}

// One wave per 16-edge tile: alpha = leaky(outi@q + outj@k + eattr*wee); atomicMax seg-max.
// QK (16 KB) staged in LDS so B-fragment loads ride the DS pipe, leaving VMEM for gathers.
__global__ void alpha_kernel(const float* __restrict__ XW, const float* __restrict__ QK,
                             const int* __restrict__ src, const int* __restrict__ dst,
                             const int* __restrict__ etype, const float* __restrict__ eattr,
                             const float* __restrict__ wee,
                             float* __restrict__ alpha, unsigned* __restrict__ amax, int E) {
  __shared__ float sQK[HC * 16];
#pragma unroll
  for (int i = 0; i < (HC * 16) / 256; ++i)
    sQK[threadIdx.x + i * 256] = QK[threadIdx.x + i * 256];
  __syncthreads();

  int wave = (int)((blockIdx.x * (long)blockDim.x + threadIdx.x) >> 5);
  int lane = threadIdx.x & 31;
  int half = lane >> 4, col = lane & 15;
  int ebase = wave * 16;
  if (ebase >= E) return;
  int e = ebase + col;                        // this lane gathers row for edge m = lane&15
  int ni = dst[e], nj = src[e], r = etype[e];
  const float* rowi = XW + ((size_t)ni * NREL + r) * HC;
  const float* rowj = XW + ((size_t)nj * NREL + r) * HC;
  v8f acc1 = {}, acc2 = {};
  for (int k0 = 0; k0 < HC; k0 += 4) {
    int ka = k0 + 2 * half;
    v2f b  = { sQK[ka * 16 + col], sQK[(ka + 1) * 16 + col] };
    v2f ai = { rowi[ka], rowi[ka + 1] };
    v2f aj = { rowj[ka], rowj[ka + 1] };
    acc1 = __builtin_amdgcn_wmma_f32_16x16x4_f32(false, ai, false, b, (short)0, acc1, false, false);
    acc2 = __builtin_amdgcn_wmma_f32_16x16x4_f32(false, aj, false, b, (short)0, acc2, false, false);
  }
  // alpha[m][h] = T1[m][h] + T2[m][h+8]: pull acc2 from lane+8 within each half
  int srcl = (lane & 16) | ((lane + 8) & 15);
#pragma unroll
  for (int v = 0; v < 8; ++v) {
    float a2 = __shfl(acc2[v], srcl, 32);     // executed by all lanes (uniform)
    if (col < HEADS) {
      int h = col;
      int mm = v + 8 * half;
      int ee = ebase + mm;
      float a = acc1[v] + a2 + eattr[ee] * wee[h];
      a = (a > 0.f) ? a : NEG_SLOPE * a;
      alpha[(size_t)ee * HEADS + h] = a;
      atomicMax(&amax[(size_t)dst[ee] * HEADS + h], f2ord(a));
    }
  }
}

// ex = exp(alpha - amax[dst]); denom[dst] += ex   (one thread per edge*head)
__global__ void softmax_ex_kernel(const int* __restrict__ dst, const unsigned* __restrict__ amax,
                                  float* __restrict__ alpha, float* __restrict__ denom, int E) {
  long t = (long)blockIdx.x * blockDim.x + threadIdx.x;
  if (t >= (long)E * HEADS) return;
  int e = (int)(t >> 3), h = (int)(t & 7);
  int d = dst[e];
  float am = ord2f(amax[(size_t)d * HEADS + h]);
  float ex = expf(alpha[t] - am);
  alpha[t] = ex;
  atomicAdd(&denom[(size_t)d * HEADS + h], ex);
}

// One wave per edge: agg[dst] += attn[h] * outj  (8 channels per lane, h = lane>>2)
__global__ void aggregate_kernel(const float* __restrict__ XW, const int* __restrict__ src,
                                 const int* __restrict__ dst, const int* __restrict__ etype,
                                 const float* __restrict__ ex, const float* __restrict__ denom,
                                 float* __restrict__ agg, int E) {
  int e = (int)((blockIdx.x * (long)blockDim.x + threadIdx.x) >> 5);
  int lane = threadIdx.x & 31;
  if (e >= E) return;
  int nj = src[e], ni = dst[e], r = etype[e];
  const float* rowj = XW + ((size_t)nj * NREL + r) * HC;
  float* aggrow = agg + (size_t)ni * HC;
  int h = lane >> 2;                          // channels [lane*8, lane*8+8) share head h
  float attn = ex[(size_t)e * HEADS + h] / (denom[(size_t)ni * HEADS + h] + 1e-16f);
  const float4* rj = (const float4*)(rowj + lane * 8);
  float4 x0 = rj[0], x1 = rj[1];
  int c = lane * 8;
  atomicAdd(&aggrow[c + 0], attn * x0.x);
  atomicAdd(&aggrow[c + 1], attn * x0.y);
  atomicAdd(&aggrow[c + 2], attn * x0.z);
  atomicAdd(&aggrow[c + 3], attn * x0.w);
  atomicAdd(&aggrow[c + 4], attn * x1.x);
  atomicAdd(&aggrow[c + 5], attn * x1.y);
  atomicAdd(&aggrow[c + 6], attn * x1.z);
  atomicAdd(&aggrow[c + 7], attn * x1.w);
}

__global__ void bias_relu_kernel(float* __restrict__ h, const float* __restrict__ b, long n) {
  long i = (long)blockIdx.x * blockDim.x + threadIdx.x;
  if (i >= n) return;
  int c = (int)(i & (HC - 1));
  float v = h[i] + b[c];
  h[i] = (v > 0.f) ? v : 0.f;
}

// out head: tanh(h @ wlin + blin) then atomic mean-pool accumulation
__global__ void head_kernel(const float* __restrict__ h, const float* __restrict__ wlin,
                            const float* __restrict__ blin, const int* __restrict__ batch,
                            float* __restrict__ sums, float* __restrict__ cnts, int N) {
  int t = blockIdx.x * blockDim.x + threadIdx.x;
  if (t >= N * HEADS) return;                 // ACT == 8 == HEADS
  int n = t >> 3, a = t & 7;
  const float* hr = h + (size_t)n * HC;
  float s = blin[a];
  for (int k = 0; k < HC; ++k) s += hr[k] * wlin[k * 8 + a];
  float v = tanhf(s);
  int g = batch[n];
  atomicAdd(&sums[g * 8 + a], v);
  if (a == 0) atomicAdd(&cnts[g], 1.0f);
}

__global__ void finalize_kernel(const float* __restrict__ sums, const float* __restrict__ cnts,
                                float* __restrict__ out, int G) {
  int t = threadIdx.x;
  if (t < G * 8) {
    int g = t >> 3;
    out[t] = sums[t] / fmaxf(cnts[g], 1.0f);
  }
}

static void run_layer(const float* hin, int K,
                      const float* w, const float* q, const float* km, const float* em,
                      const float* we, const float* b,
                      const int* src, const int* dst, const int* ety, const float* eattr,
                      float* xw, float* QK, float* wee, float* alpha, unsigned* amax,
                      float* denom, float* hout, int N, int E, hipStream_t stream) {
  build_qk_wee_kernel<<<(HC * 16 + 255) / 256, 256, 0, stream>>>(q, km, em, we, QK, wee);
  fill_u32_kernel<<<(N * 8 + 255) / 256, 256, 0, stream>>>(amax, 0u, (long)N * 8);
  fill_u32_kernel<<<(N * 8 + 255) / 256, 256, 0, stream>>>((unsigned*)denom, 0u, (long)N * 8);
  fill_u32_kernel<<<((long)N * HC + 255) / 256, 256, 0, stream>>>((unsigned*)hout, 0u, (long)N * HC);
  dim3 g(N / 16, HC / 64, NREL);
  gemm_xw_kernel<<<g, 32, 0, stream>>>(hin, w, xw, K);
  long alpha_threads = (long)(E / 16) * 32;
  alpha_kernel<<<(alpha_threads + 255) / 256, 256, 0, stream>>>(xw, QK, src, dst, ety, eattr,
                                                                wee, alpha, amax, E);
  softmax_ex_kernel<<<((long)E * 8 + 255) / 256, 256, 0, stream>>>(dst, amax, alpha, denom, E);
  aggregate_kernel<<<((long)E * 32 + 255) / 256, 256, 0, stream>>>(xw, src, dst, ety, alpha,
                                                                   denom, hout, E);
  bias_relu_kernel<<<((long)N * HC + 255) / 256, 256, 0, stream>>>(hout, b, (long)N * HC);
}

extern "C" void kernel_launch(void* const* d_in, const int* in_sizes, int n_in,
                              void* d_out, int out_size, void* d_ws, size_t ws_size,
                              hipStream_t stream) {
  (void)n_in; (void)ws_size;
  const float* x      = (const float*)d_in[0];
  const int* eidx     = (const int*)d_in[1];
  const int* ety      = (const int*)d_in[2];
  const float* eattr  = (const float*)d_in[3];
  const int* batch    = (const int*)d_in[4];
  const float* p[20];
  for (int i = 0; i < 18; ++i) p[i] = (const float*)d_in[5 + i];   // 3 layers x (w,q,k,e,we,b)
  const float* wlin = (const float*)d_in[23];
  const float* blin = (const float*)d_in[24];

  int N = in_sizes[0] / 32;        // OBS = 32
  int E = in_sizes[1] / 2;
  int G = out_size / 8;            // ACT = 8
  const int* src = eidx;
  const int* dst = eidx + E;

  // workspace carve (floats)
  float* ws    = (float*)d_ws;
  float* xw    = ws;                               // N*3*HC
  float* hA    = xw + (size_t)N * NREL * HC;       // N*HC
  float* hB    = hA + (size_t)N * HC;              // N*HC
  float* alpha = hB + (size_t)N * HC;              // E*8
  unsigned* amax = (unsigned*)(alpha + (size_t)E * 8);  // N*8
  float* denom = (float*)(amax + (size_t)N * 8);   // N*8
  float* QK    = denom + (size_t)N * 8;            // HC*16
  float* wee   = QK + HC * 16;                     // 8
  float* sums  = wee + 8;                          // G*8
  float* cnts  = sums + (size_t)G * 8;             // G

  run_layer(x,  32, p[0],  p[1],  p[2],  p[3],  p[4],  p[5],  src, dst, ety, eattr,
            xw, QK, wee, alpha, amax, denom, hA, N, E, stream);
  run_layer(hA, HC, p[6],  p[7],  p[8],  p[9],  p[10], p[11], src, dst, ety, eattr,
            xw, QK, wee, alpha, amax, denom, hB, N, E, stream);
  run_layer(hB, HC, p[12], p[13], p[14], p[15], p[16], p[17], src, dst, ety, eattr,
            xw, QK, wee, alpha, amax, denom, hA, N, E, stream);

  fill_u32_kernel<<<1, 256, 0, stream>>>((unsigned*)sums, 0u, (long)G * 9);  // sums + cnts
  head_kernel<<<(N * 8 + 255) / 256, 256, 0, stream>>>(hA, wlin, blin, batch, sums, cnts, N);
  finalize_kernel<<<1, 256, 0, stream>>>(sums, cnts, (float*)d_out, G);
}